// EmbeddingModule_45792941310185
// MI455X (gfx1250) — compile-verified
//
#include <hip/hip_runtime.h>
#include <hip/hip_bf16.h>

#define DIMD 128
#define BINK 16
#define LDN  50
#define LCN  50
#define LTOT 100
#define OUTO 64
#define LN_EPS 1e-5f

#define PSTRIDE 136   // padded row stride (floats) for parts in LDS
#define TSTRIDE 101   // padded row stride (floats) for T in LDS

typedef __attribute__((ext_vector_type(2))) float v2f;
typedef __attribute__((ext_vector_type(8))) float v8f;

__global__ __launch_bounds__(256)
void fused_embed_transform(
    const int*   __restrict__ discrete_index,
    const int*   __restrict__ discrete_mask,
    const float* __restrict__ continue_value,
    const int*   __restrict__ continue_index,
    const int*   __restrict__ continue_mask,
    const float* __restrict__ disc_emb,
    const float* __restrict__ v_emb,
    const float* __restrict__ b1_emb,
    const float* __restrict__ w_emb,
    const float* __restrict__ b2_emb,
    const float* __restrict__ ln_gamma,
    const float* __restrict__ ln_beta,
    const float* __restrict__ Tm,
    float* __restrict__ out,        // [B, 64, 128] fp32
    float* __restrict__ mask_out,   // [B, 64] written as float
    int B)
{
    __shared__ float parts[LTOT * PSTRIDE];   // 54.4 KB
    __shared__ float Tlds[OUTO * TSTRIDE];    // 25.9 KB
    __shared__ float a_buf[LCN * BINK];       //  3.2 KB
    __shared__ float mv[LTOT];
    __shared__ int   cont_idx[LCN];
    __shared__ int   disc_idx[LDN];

    const int b   = blockIdx.x;
    const int tid = threadIdx.x;

    // ---- stage T into LDS (padded rows) ----
    for (int i = tid; i < OUTO * LTOT; i += 256) {
        int r = i / LTOT, c = i - r * LTOT;
        Tlds[r * TSTRIDE + c] = Tm[i];
    }

    // ---- stage indices/masks; per-token LN+ReLU front half ----
    if (tid < LCN) {
        const int l = tid;
        const int idx = continue_index[(size_t)b * LCN + l];
        cont_idx[l] = idx;
        const float cv = continue_value[(size_t)b * LCN + l];
        const float4* vr  = (const float4*)(v_emb  + (size_t)idx * BINK);
        const float4* b1r = (const float4*)(b1_emb + (size_t)idx * BINK);
        float h[BINK];
        #pragma unroll
        for (int q = 0; q < 4; ++q) {
            float4 v4 = vr[q], b4 = b1r[q];
            h[4*q+0] = cv*v4.x + b4.x; h[4*q+1] = cv*v4.y + b4.y;
            h[4*q+2] = cv*v4.z + b4.z; h[4*q+3] = cv*v4.w + b4.w;
        }
        float mu = 0.f;
        #pragma unroll
        for (int k = 0; k < BINK; ++k) mu += h[k];
        mu *= (1.0f / BINK);
        float var = 0.f;
        #pragma unroll
        for (int k = 0; k < BINK; ++k) { float d = h[k] - mu; var += d * d; }
        var *= (1.0f / BINK);
        const float rs = rsqrtf(var + LN_EPS);
        #pragma unroll
        for (int k = 0; k < BINK; ++k) {
            float hn = (h[k] - mu) * rs * ln_gamma[k] + ln_beta[k];
            a_buf[l * BINK + k] = hn > 0.f ? hn : 0.f;
        }
    } else if (tid >= 64 && tid < 64 + LTOT) {
        const int l = tid - 64;
        const int m = (l < LDN) ? discrete_mask[(size_t)b * LDN + l]
                                : continue_mask[(size_t)b * LCN + (l - LDN)];
        mv[l] = 1.0f - (float)m;
    } else if (tid >= 192 && tid < 192 + LDN) {
        const int l = tid - 192;
        disc_idx[l] = discrete_index[(size_t)b * LDN + l];
    }
    __syncthreads();

    // ---- fill parts (LDS): discrete gather + continuous per-token bmm ----
    for (int c = tid; c < 3200; c += 256) {
        if (c < 1600) {
            const int l = c >> 5;
            const int j = c & 31;
            const int row = disc_idx[l];
            float4 v4 = ((const float4*)(disc_emb + (size_t)row * DIMD))[j];
            *(float4*)(&parts[l * PSTRIDE + j * 4]) = v4;
        } else {
            const int c2 = c - 1600;
            const int l = c2 >> 5;
            const int j = c2 & 31;
            const int idx = cont_idx[l];
            const float4* wr = (const float4*)(w_emb + (size_t)idx * (DIMD * BINK));
            float4 acc = ((const float4*)(b2_emb + (size_t)idx * DIMD))[j];
            #pragma unroll
            for (int k = 0; k < BINK; ++k) {
                const float ak = a_buf[l * BINK + k];
                float4 wv = wr[k * 32 + j];
                acc.x += ak * wv.x; acc.y += ak * wv.y;
                acc.z += ak * wv.z; acc.w += ak * wv.w;
            }
            *(float4*)(&parts[(LDN + l) * PSTRIDE + j * 4]) = acc;
        }
    }

    // ---- mask output: full_mask[b,o] = 1 - sign(T[o,:] . mv) ----
    if (tid < OUTO) {
        float s = 0.f;
        for (int l = 0; l < LTOT; ++l) s += Tlds[tid * TSTRIDE + l] * mv[l];
        const float sg = (s > 0.f) ? 1.f : ((s < 0.f) ? -1.f : 0.f);
        mask_out[(size_t)b * OUTO + tid] = 1.0f - sg;
    }

    __syncthreads();

    // ---- WMMA: out[b](64x128) = T(64x100) x parts(100x128), V_WMMA_F32_16X16X4_F32 ----
    const int wave  = tid >> 5;
    const int lane  = tid & 31;
    const int mo    = wave & 3;          // M-tile 0..3 (rows mo*16..mo*16+15)
    const int ng    = (wave >> 2) * 4;   // N-tile group base: 0 or 4
    const int lhalf = lane >> 4;         // 0 = lanes 0-15, 1 = lanes 16-31
    const int l16   = lane & 15;

    v8f acc0 = {}, acc1 = {}, acc2 = {}, acc3 = {};

    const int arow = mo * 16 + l16;
    for (int ks = 0; ks < 25; ++ks) {
        // A 16x4 f32 layout: lanes0-15 hold K=kb,kb+1; lanes16-31 hold K=kb+2,kb+3
        const int kb = ks * 4 + lhalf * 2;
        v2f A;
        A.x = Tlds[arow * TSTRIDE + kb];
        A.y = Tlds[arow * TSTRIDE + kb + 1];
        #pragma unroll
        for (int j = 0; j < 4; ++j) {
            const int n = (ng + j) * 16 + l16;
            v2f Bv;
            Bv.x = parts[(kb    ) * PSTRIDE + n];
            Bv.y = parts[(kb + 1) * PSTRIDE + n];
            v8f* ap = (j == 0) ? &acc0 : (j == 1) ? &acc1 : (j == 2) ? &acc2 : &acc3;
            *ap = __builtin_amdgcn_wmma_f32_16x16x4_f32(
                /*neg_a=*/false, A, /*neg_b=*/false, Bv,
                /*c_mod=*/(short)0, *ap, /*reuse_a=*/false, /*reuse_b=*/false);
        }
    }

    // ---- store D tiles: VGPR r -> M = mo*16 + r + lhalf*8, N = tile*16 + l16 ----
    float* ob = out + (size_t)b * (OUTO * DIMD);
    #pragma unroll
    for (int j = 0; j < 4; ++j) {
        v8f accv = (j == 0) ? acc0 : (j == 1) ? acc1 : (j == 2) ? acc2 : acc3;
        const int n = (ng + j) * 16 + l16;
        #pragma unroll
        for (int r = 0; r < 8; ++r) {
            const int m = mo * 16 + r + lhalf * 8;
            ob[m * DIMD + n] = accv[r];
        }
    }
}

extern "C" void kernel_launch(void* const* d_in, const int* in_sizes, int n_in,
                              void* d_out, int out_size, void* d_ws, size_t ws_size,
                              hipStream_t stream) {
    const int*   discrete_index = (const int*)  d_in[0];
    const int*   discrete_mask  = (const int*)  d_in[1];
    const float* continue_value = (const float*)d_in[2];
    const int*   continue_index = (const int*)  d_in[3];
    const int*   continue_mask  = (const int*)  d_in[4];
    const float* disc_emb       = (const float*)d_in[5];
    const float* v_emb          = (const float*)d_in[6];
    const float* b1_emb         = (const float*)d_in[7];
    const float* w_emb          = (const float*)d_in[8];
    const float* b2_emb         = (const float*)d_in[9];
    const float* ln_gamma       = (const float*)d_in[10];
    const float* ln_beta        = (const float*)d_in[11];
    const float* Tm             = (const float*)d_in[12];

    const int B = in_sizes[0] / LDN;   // 4096

    float* out      = (float*)d_out;
    float* mask_out = out + (size_t)B * OUTO * DIMD;

    fused_embed_transform<<<dim3(B), dim3(256), 0, stream>>>(
        discrete_index, discrete_mask, continue_value, continue_index, continue_mask,
        disc_emb, v_emb, b1_emb, w_emb, b2_emb, ln_gamma, ln_beta, Tm,
        out, mask_out, B);
}